// Sampler2D_27247272526493
// MI455X (gfx1250) — compile-verified
//
#include <hip/hip_runtime.h>

// Bilinear 2D sampler for MI455X (gfx1250), round 3.
// L2-gather-bound: 100.7 MB fp16 image resides in the 192 MB L2 (23.3 TB/s
// HBM never touched after warm-up). Round-2 asm confirmed 12-byte row reads
// lower to single wide loads and coords/outputs are b128 NT streams.
// Round-3 change: split each thread's work into explicit phases
//   (1) address+weight setup for all 4 samples,
//   (2) all 8 independent row gathers issued back-to-back (one load clause),
//   (3) f32 accumulation,
// so all texel gathers are in flight before the first s_wait_loadcnt ->
// maximum per-wave memory-level parallelism for L2 latency hiding.
// WMMA not applicable: per-sample weights AND per-sample gathered data.

typedef float v4f __attribute__((ext_vector_type(4)));

static constexpr int kW   = 4096;
static constexpr int kH   = 4096;
static constexpr int kC   = 3;
static constexpr int kIPT = 4;     // samples per thread

// Phase 1: compute row pointers + interpolation weights for one sample.
__device__ __forceinline__ void setup_one(float cx, float cy,
                                          const _Float16* __restrict__ data,
                                          const _Float16** r0,
                                          const _Float16** r1,
                                          bool* edge, float w[4]) {
    // xs = x * [W, H]; ind = trunc(xs); frac from UNclamped ind (ref semantics)
    float fx = cx * (float)kW;
    float fy = cy * (float)kH;
    int ix = (int)fx;
    int iy = (int)fy;
    float wx = fx - (float)ix;
    float wy = fy - (float)iy;

    int x0 = min(max(ix, 0), kW - 1);
    int y0 = min(max(iy, 0), kH - 1);
    int y1 = min(y0 + 1, kH - 1);

    bool e  = (x0 == kW - 1);            // x1 == x0 only here
    int  xa = e ? (kW - 2) : x0;         // 12B row read covers [xa, xa+1]

    *r0 = data + ((size_t)((y0 << 12) + xa) * kC);
    *r1 = data + ((size_t)((y1 << 12) + xa) * kC);
    *edge = e;

    w[0] = (1.0f - wx) * (1.0f - wy);    // w00
    w[1] = wx * (1.0f - wy);             // w01
    w[2] = (1.0f - wx) * wy;             // w10
    w[3] = wx * wy;                      // w11
}

// Phase 3: 4-tap weighted sum for one sample from its two fetched rows.
__device__ __forceinline__ void accum_one(const _Float16 h0[6],
                                          const _Float16 h1[6],
                                          bool edge, const float w[4],
                                          float r[kC]) {
#pragma unroll
    for (int c = 0; c < kC; ++c) {
        // edge: texel(W-1) sits in halves [3..5]; it is both x0 and x1.
        float t00 = edge ? (float)h0[c + 3] : (float)h0[c];
        float t01 = (float)h0[c + 3];
        float t10 = edge ? (float)h1[c + 3] : (float)h1[c];
        float t11 = (float)h1[c + 3];
        r[c] = t00 * w[0] + t01 * w[1] + t10 * w[2] + t11 * w[3];
    }
}

__global__ __launch_bounds__(256) void sampler2d_kernel(
    const float* __restrict__ x,        // (N,2) f32 coords in [0,1)
    const _Float16* __restrict__ data,  // (H,W,C) f16 image
    float* __restrict__ out,            // (N,3) f32
    int n) {
    int tid = blockIdx.x * blockDim.x + threadIdx.x;
    long long base = (long long)tid * kIPT;
    if (base >= n) return;

    if (base + kIPT <= n) {
        // ---- coord stream: 2x b128 NT loads (keep L2 for the image) ----
        const v4f* __restrict__ xv = reinterpret_cast<const v4f*>(x);
        v4f c01 = __builtin_nontemporal_load(xv + 2 * (size_t)tid);     // s0,s1
        v4f c23 = __builtin_nontemporal_load(xv + 2 * (size_t)tid + 1); // s2,s3

        const float cx[kIPT] = { c01.x, c01.z, c23.x, c23.z };
        const float cy[kIPT] = { c01.y, c01.w, c23.y, c23.w };

        // ---- phase 1: all addresses + weights ----
        const _Float16* p[2 * kIPT];
        bool  edge[kIPT];
        float w[kIPT][4];
#pragma unroll
        for (int s = 0; s < kIPT; ++s)
            setup_one(cx[s], cy[s], data, &p[2 * s], &p[2 * s + 1],
                      &edge[s], w[s]);

        // ---- phase 2: 8 independent 12B row gathers, back-to-back ----
        _Float16 h[2 * kIPT][6];
#pragma unroll
        for (int k = 0; k < 2 * kIPT; ++k)
            __builtin_memcpy(h[k], p[k], 12);

        // ---- phase 3: accumulate in f32 ----
        float r[kIPT * kC];
#pragma unroll
        for (int s = 0; s < kIPT; ++s)
            accum_one(h[2 * s], h[2 * s + 1], edge[s], w[s], r + s * kC);

        // ---- output stream: 3x b128 NT stores (48B, 16B-aligned) ----
        v4f o0 = { r[0], r[1], r[2],  r[3] };
        v4f o1 = { r[4], r[5], r[6],  r[7] };
        v4f o2 = { r[8], r[9], r[10], r[11] };
        v4f* ov = reinterpret_cast<v4f*>(out + (size_t)base * kC);
        __builtin_nontemporal_store(o0, ov + 0);
        __builtin_nontemporal_store(o1, ov + 1);
        __builtin_nontemporal_store(o2, ov + 2);
    } else {
        // Tail path (not taken for N = 4194304, kept for generality).
        for (long long i = base; i < n; ++i) {
            const _Float16 *r0, *r1;
            bool  e;
            float w[4];
            setup_one(x[2 * i], x[2 * i + 1], data, &r0, &r1, &e, w);
            _Float16 h0[6], h1[6];
            __builtin_memcpy(h0, r0, 12);
            __builtin_memcpy(h1, r1, 12);
            float r[kC];
            accum_one(h0, h1, e, w, r);
            out[3 * i + 0] = r[0];
            out[3 * i + 1] = r[1];
            out[3 * i + 2] = r[2];
        }
    }
}

extern "C" void kernel_launch(void* const* d_in, const int* in_sizes, int n_in,
                              void* d_out, int out_size, void* d_ws, size_t ws_size,
                              hipStream_t stream) {
    const float*    x    = (const float*)d_in[0];      // (N,2) float32
    const _Float16* data = (const _Float16*)d_in[1];   // (H,W,C) float16
    float*          out  = (float*)d_out;              // (N,3) float32

    int n = in_sizes[0] / 2;                           // N samples
    int threads_total = (n + kIPT - 1) / kIPT;         // 4 samples per thread
    int block = 256;                                   // 8 wave32 per block
    int grid = (threads_total + block - 1) / block;

    sampler2d_kernel<<<grid, block, 0, stream>>>(x, data, out, n);
}